// MOE_5686536700148
// MI455X (gfx1250) — compile-verified
//
#include <hip/hip_runtime.h>
#include <math.h>

// ---------------- problem constants ----------------
#define T_TOK 1024
#define DIM   1536
#define IDIM  4608
#define NEXP  8
#define TOPK  2
#define TR    (T_TOK * TOPK)   // 2048 token-expert pairs
#define MCHUNK 256             // rows per workgroup chunk (16 WMMA M-subtiles)

typedef __attribute__((ext_vector_type(2))) float v2f;
typedef __attribute__((ext_vector_type(8))) float v8f;

// fp32 dense WMMA: D(16x16 f32) = A(16x4 f32) * B(4x16 f32) + C
__device__ __forceinline__ v8f wmma_f32(v2f a, v2f b, v8f c) {
  return __builtin_amdgcn_wmma_f32_16x16x4_f32(false, a, false, b, (short)0, c,
                                               false, false);
}

// ---------------- kernel 0: zero counts ----------------
__global__ void moe_init(int* __restrict__ counts) {
  if (threadIdx.x < NEXP && blockIdx.x == 0) counts[threadIdx.x] = 0;
}

// ---------------- kernel 1: router ----------------
__global__ void moe_router(const float* __restrict__ x,
                           const float* __restrict__ rw,      // [DIM, NEXP]
                           float* __restrict__ probs,         // [T, NEXP]
                           float* __restrict__ zsq,           // [T]
                           int*   __restrict__ topk_idx,      // [T, 2]
                           float* __restrict__ topk_w,        // [T, 2]
                           int*   __restrict__ counts) {      // [NEXP]
  int t = blockIdx.x * blockDim.x + threadIdx.x;
  if (t >= T_TOK) return;
  float l[NEXP];
#pragma unroll
  for (int e = 0; e < NEXP; ++e) l[e] = 0.f;
  const float* xr = x + (size_t)t * DIM;
  const float4* rw4 = (const float4*)rw;
  for (int d = 0; d < DIM; ++d) {
    float xv = xr[d];
    float4 r0 = rw4[d * 2], r1 = rw4[d * 2 + 1];
    l[0] += xv * r0.x; l[1] += xv * r0.y; l[2] += xv * r0.z; l[3] += xv * r0.w;
    l[4] += xv * r1.x; l[5] += xv * r1.y; l[6] += xv * r1.z; l[7] += xv * r1.w;
  }
  float m = l[0];
#pragma unroll
  for (int e = 1; e < NEXP; ++e) m = fmaxf(m, l[e]);
  float p[NEXP], s = 0.f;
#pragma unroll
  for (int e = 0; e < NEXP; ++e) { p[e] = expf(l[e] - m); s += p[e]; }
  float inv = 1.f / s;
#pragma unroll
  for (int e = 0; e < NEXP; ++e) { p[e] *= inv; probs[t * NEXP + e] = p[e]; }
  float z = m + logf(s);
  zsq[t] = z * z;

  int i1 = 0;
#pragma unroll
  for (int e = 1; e < NEXP; ++e) if (l[e] > l[i1]) i1 = e;   // strict >: lowest idx wins ties
  int i2 = (i1 == 0) ? 1 : 0;
#pragma unroll
  for (int e = 0; e < NEXP; ++e) if (e != i1 && l[e] > l[i2]) i2 = e;

  float wsum = p[i1] + p[i2];
  topk_idx[t * 2 + 0] = i1;
  topk_idx[t * 2 + 1] = i2;
  topk_w[t * 2 + 0] = p[i1] / wsum;
  topk_w[t * 2 + 1] = p[i2] / wsum;
  atomicAdd(&counts[i1], 1);   // integer atomics: deterministic
  atomicAdd(&counts[i2], 1);
}

// ---------------- kernel 2: deterministic router reductions ----------------
__global__ void moe_router_reduce(const float* __restrict__ probs,
                                  const float* __restrict__ zsq,
                                  float* __restrict__ p_mean,   // [NEXP]
                                  float* __restrict__ zmean) {  // [1]
  __shared__ float red[256];
  int t = threadIdx.x;
  for (int j = 0; j < NEXP + 1; ++j) {
    float s = 0.f;
    for (int i = t; i < T_TOK; i += 256)
      s += (j < NEXP) ? probs[i * NEXP + j] : zsq[i];
    red[t] = s;
    __syncthreads();
    for (int k = 128; k > 0; k >>= 1) {
      if (t < k) red[t] += red[t + k];
      __syncthreads();
    }
    if (t == 0) {
      if (j < NEXP) p_mean[j] = red[0] / (float)T_TOK;
      else          zmean[0]  = red[0] / (float)T_TOK;
    }
    __syncthreads();
  }
}

// ---------------- kernel 3: stable permutation ----------------
__global__ void moe_permute(const int* __restrict__ counts,
                            const int* __restrict__ topk_idx,   // flat [TR]
                            int* __restrict__ offsets,          // [NEXP+1]
                            int* __restrict__ row_map,          // [TR]
                            int* __restrict__ pos_of_pair) {    // [TR]
  if (blockIdx.x != 0 || threadIdx.x != 0) return;
  int cur[NEXP];
  int off = 0;
#pragma unroll
  for (int e = 0; e < NEXP; ++e) { offsets[e] = off; cur[e] = off; off += counts[e]; }
  offsets[NEXP] = off;
  for (int i = 0; i < TR; ++i) {   // stable within expert bucket
    int e = topk_idx[i];
    int pos = cur[e]++;
    row_map[pos] = i;
    pos_of_pair[i] = pos;
  }
}

// ---------------- grouped WMMA GEMM ----------------
// WG: expert e (blockIdx.y), 64 output cols (4 waves x 16), 256-row chunk (blockIdx.z).
// LDS double buffer, row-major stride-20 so each A fragment is one aligned ds_load_b64
// landing directly in the even VGPR pair the WMMA consumes (no repack movs).
// B (weights) streamed straight from HBM -> read ~once per expert (roofline floor).
template <int DIN, int DOUT, bool GATE>
__global__ __launch_bounds__(128)
void moe_grouped_gemm(const float* __restrict__ A,    // GATE: x [T,DIN]; else gated [TR,DIN]
                      const float* __restrict__ W0,   // [NEXP, DIN, DOUT]
                      const float* __restrict__ W1,   // [NEXP, DIN, DOUT] or null
                      const int* __restrict__ offsets,
                      const int* __restrict__ row_map,
                      float* __restrict__ out) {      // [TR, DOUT]
  __shared__ float lx[2][MCHUNK][20];   // 40 KB of 320 KB WGP LDS

  const int e  = blockIdx.y;
  const int g0 = offsets[e];
  const int g1 = offsets[e + 1];
  const int r_begin = g0 + blockIdx.z * MCHUNK;
  if (r_begin >= g1) return;
  const int rows = min(MCHUNK, g1 - r_begin);

  const int tid  = threadIdx.x;
  const int lane = tid & 31;
  const int wave = tid >> 5;
  const int n0   = blockIdx.x * 64 + wave * 16;

  // each thread stages two rows (tid, tid+128); padded rows stage zeros
  const float* arow0 = nullptr;
  const float* arow1 = nullptr;
  if (tid < rows) {
    int src = GATE ? (row_map[r_begin + tid] / TOPK) : (r_begin + tid);
    arow0 = A + (size_t)src * DIN;
  }
  if (tid + 128 < rows) {
    int src = GATE ? (row_map[r_begin + tid + 128] / TOPK) : (r_begin + tid + 128);
    arow1 = A + (size_t)src * DIN;
  }

  const float* wb0 = W0 + (size_t)e * DIN * DOUT + n0 + (lane & 15);
  const float* wb1 = GATE ? (W1 + (size_t)e * DIN * DOUT + n0 + (lane & 15)) : nullptr;

  const v8f vz = {0.f, 0.f, 0.f, 0.f, 0.f, 0.f, 0.f, 0.f};
  v8f acc0[16], acc1[16];
#pragma unroll
  for (int m = 0; m < 16; ++m) { acc0[m] = vz; if (GATE) acc1[m] = vz; }

  const int khalf = (lane & 16) ? 2 : 0;   // half-wave K offset per fp32 WMMA layout
  const int arl   = lane & 15;

  float4 rg[8];
  auto fetch = [&](int kb) {
    const float4 z4 = make_float4(0.f, 0.f, 0.f, 0.f);
    if (arow0) {
      const float4* s = (const float4*)(arow0 + kb);
      rg[0] = s[0]; rg[1] = s[1]; rg[2] = s[2]; rg[3] = s[3];
    } else { rg[0] = rg[1] = rg[2] = rg[3] = z4; }
    if (arow1) {
      const float4* s = (const float4*)(arow1 + kb);
      rg[4] = s[0]; rg[5] = s[1]; rg[6] = s[2]; rg[7] = s[3];
    } else { rg[4] = rg[5] = rg[6] = rg[7] = z4; }
  };
  auto stage = [&](int buf) {
    float2* d0 = (float2*)&lx[buf][tid][0];          // 8B aligned (stride 80B)
    d0[0] = make_float2(rg[0].x, rg[0].y); d0[1] = make_float2(rg[0].z, rg[0].w);
    d0[2] = make_float2(rg[1].x, rg[1].y); d0[3] = make_float2(rg[1].z, rg[1].w);
    d0[4] = make_float2(rg[2].x, rg[2].y); d0[5] = make_float2(rg[2].z, rg[2].w);
    d0[6] = make_float2(rg[3].x, rg[3].y); d0[7] = make_float2(rg[3].z, rg[3].w);
    float2* d1 = (float2*)&lx[buf][tid + 128][0];
    d1[0] = make_float2(rg[4].x, rg[4].y); d1[1] = make_float2(rg[4].z, rg[4].w);
    d1[2] = make_float2(rg[5].x, rg[5].y); d1[3] = make_float2(rg[5].z, rg[5].w);
    d1[4] = make_float2(rg[6].x, rg[6].y); d1[5] = make_float2(rg[6].z, rg[6].w);
    d1[6] = make_float2(rg[7].x, rg[7].y); d1[7] = make_float2(rg[7].z, rg[7].w);
  };

  const int nk = DIN / 16;
  fetch(0);
  stage(0);
  __syncthreads();

  for (int ib = 0; ib < nk; ++ib) {
    const int kb  = ib * 16;
    const int buf = ib & 1;
    if (ib + 1 < nk) fetch(kb + 16);

    const float(*L)[20] = lx[buf];
#pragma unroll
    for (int k4 = 0; k4 < 16; k4 += 4) {
      const int kr = kb + k4 + khalf;
      // B 4x16 fp32: VGPR0 lanes0-15 row K, lanes16-31 row K+2; VGPR1 rows K+1/K+3
      v2f b0; b0.x = wb0[(size_t)kr * DOUT]; b0.y = wb0[(size_t)(kr + 1) * DOUT];
      v2f b1;
      if (GATE) { b1.x = wb1[(size_t)kr * DOUT]; b1.y = wb1[(size_t)(kr + 1) * DOUT]; }
      const int ac = k4 + khalf;
#pragma unroll
      for (int m = 0; m < 16; ++m) {
        // contiguous pair -> single aligned ds_load_b64 into even VGPR pair
        float2 af = *(const float2*)&L[m * 16 + arl][ac];
        v2f a; a.x = af.x; a.y = af.y;
        acc0[m] = wmma_f32(a, b0, acc0[m]);
        if (GATE) acc1[m] = wmma_f32(a, b1, acc1[m]);
      }
    }

    if (ib + 1 < nk) stage(1 - buf);   // write other buffer; one barrier per block
    __syncthreads();
  }

  // epilogue: C/D layout — VGPR v: lanes0-15 M=v, lanes16-31 M=v+8; N = lane%16
  const int col   = n0 + (lane & 15);
  const int rhalf = (lane & 16) ? 8 : 0;
#pragma unroll
  for (int m = 0; m < 16; ++m) {
#pragma unroll
    for (int v = 0; v < 8; ++v) {
      int lr = m * 16 + v + rhalf;
      if (lr < rows) {
        size_t r = (size_t)(r_begin + lr);
        if (GATE) {
          float g = acc0[m][v], u = acc1[m][v];
          float sg = g / (1.f + expf(-g));      // silu
          out[r * DOUT + col] = sg * u;
        } else {
          out[r * DOUT + col] = acc0[m][v];
        }
      }
    }
  }
}

// ---------------- kernel 6: weighted combine (unpermute) ----------------
__global__ void moe_combine(const float* __restrict__ out_sorted,  // [TR, DIM]
                            const int* __restrict__ pos_of_pair,   // [TR]
                            const float* __restrict__ topk_w,      // [T, 2]
                            float* __restrict__ out) {             // [T, DIM]
  int i = blockIdx.x * blockDim.x + threadIdx.x;   // over T*DIM/4
  if (i >= T_TOK * DIM / 4) return;
  int t  = i / (DIM / 4);
  int d4 = i % (DIM / 4);
  const float4* s0 = (const float4*)(out_sorted + (size_t)pos_of_pair[t * 2 + 0] * DIM);
  const float4* s1 = (const float4*)(out_sorted + (size_t)pos_of_pair[t * 2 + 1] * DIM);
  float w0 = topk_w[t * 2 + 0], w1 = topk_w[t * 2 + 1];
  float4 a = s0[d4], b = s1[d4];
  float4 o;
  o.x = w0 * a.x + w1 * b.x;
  o.y = w0 * a.y + w1 * b.y;
  o.z = w0 * a.z + w1 * b.z;
  o.w = w0 * a.w + w1 * b.w;
  ((float4*)out)[i] = o;
}

// ---------------- kernel 7: aux losses ----------------
__global__ void moe_losses(const int* __restrict__ counts,
                           const float* __restrict__ p_mean,
                           const float* __restrict__ zmean,
                           float* __restrict__ out_tail) {  // d_out + T*DIM
  if (blockIdx.x != 0 || threadIdx.x != 0) return;
  float lbl = 0.f;
#pragma unroll
  for (int e = 0; e < NEXP; ++e) {
    float load = (float)counts[e] / (float)TR;
    out_tail[e] = load;
    lbl += load * ((float)NEXP / (float)TOPK) * p_mean[e];
  }
  out_tail[NEXP]     = 0.01f  * lbl;      // lbl
  out_tail[NEXP + 1] = 0.001f * zmean[0]; // rzl
}

// ---------------- host launcher ----------------
extern "C" void kernel_launch(void* const* d_in, const int* in_sizes, int n_in,
                              void* d_out, int out_size, void* d_ws, size_t ws_size,
                              hipStream_t stream) {
  const float* x        = (const float*)d_in[0];
  const float* router_w = (const float*)d_in[1];
  const float* w1       = (const float*)d_in[2];
  const float* w2       = (const float*)d_in[3];
  const float* w3       = (const float*)d_in[4];
  float* out = (float*)d_out;

  // workspace carve (~51 MB total)
  char* p = (char*)d_ws;
  auto carve = [&](size_t bytes) -> void* {
    void* r = (void*)p;
    p += (bytes + 255) & ~(size_t)255;
    return r;
  };
  float* probs       = (float*)carve((size_t)T_TOK * NEXP * 4);
  float* zsq         = (float*)carve((size_t)T_TOK * 4);
  int*   topk_idx    = (int*)  carve((size_t)T_TOK * TOPK * 4);
  float* topk_w      = (float*)carve((size_t)T_TOK * TOPK * 4);
  int*   counts      = (int*)  carve(NEXP * 4);
  int*   offsets     = (int*)  carve((NEXP + 1) * 4);
  int*   row_map     = (int*)  carve((size_t)TR * 4);
  int*   pos_of_pair = (int*)  carve((size_t)TR * 4);
  float* p_mean      = (float*)carve(NEXP * 4);
  float* zmean       = (float*)carve(4);
  float* gated       = (float*)carve((size_t)TR * IDIM * 4);   // 37.75 MB
  float* out_sorted  = (float*)carve((size_t)TR * DIM * 4);    // 12.58 MB

  moe_init<<<1, 32, 0, stream>>>(counts);

  moe_router<<<T_TOK / 256, 256, 0, stream>>>(x, router_w, probs, zsq,
                                              topk_idx, topk_w, counts);

  moe_router_reduce<<<1, 256, 0, stream>>>(probs, zsq, p_mean, zmean);

  moe_permute<<<1, 1, 0, stream>>>(counts, topk_idx, offsets, row_map, pos_of_pair);

  // fused w1/w2 + SiLU grouped GEMM: [rows,1536] x [1536,4608] (x2) -> gated
  {
    dim3 grid(IDIM / 64, NEXP, TR / MCHUNK);
    moe_grouped_gemm<DIM, IDIM, true><<<grid, 128, 0, stream>>>(
        x, w1, w2, offsets, row_map, gated);
  }
  // w3 grouped GEMM: [rows,4608] x [4608,1536] -> out_sorted
  {
    dim3 grid(DIM / 64, NEXP, TR / MCHUNK);
    moe_grouped_gemm<IDIM, DIM, false><<<grid, 128, 0, stream>>>(
        gated, w3, nullptr, offsets, row_map, out_sorted);
  }

  moe_combine<<<(T_TOK * DIM / 4 + 255) / 256, 256, 0, stream>>>(
      out_sorted, pos_of_pair, topk_w, out);

  moe_losses<<<1, 32, 0, stream>>>(counts, p_mean, zmean, out + (size_t)T_TOK * DIM);
}